// GraphSAGELayer_71743133712859
// MI455X (gfx1250) — compile-verified
//
#include <hip/hip_runtime.h>
#include <hip/hip_bf16.h>

#define N_NODES 50000
#define N_EDGES 800000
#define DIM 64

typedef float v2f __attribute__((ext_vector_type(2)));
typedef float v8f __attribute__((ext_vector_type(8)));

// ---------------------------------------------------------------------------
// Kernel 0: zero the neigh accumulator (d_ws is poisoned before timing)
// ---------------------------------------------------------------------------
__global__ void sage_zero_ws(float4* __restrict__ ws, int n4) {
    int i = blockIdx.x * blockDim.x + threadIdx.x;
    if (i < n4) ws[i] = make_float4(0.f, 0.f, 0.f, 0.f);
}

// ---------------------------------------------------------------------------
// Kernel 1: msgs = edge_w * h[src]; neigh[dst] += msgs  (atomic scatter-add)
// 16 lanes per edge, each lane handles 4 consecutive features (float4 gather,
// 4x global_atomic_add_f32). Whole working set lives in L2 (192 MB).
// ---------------------------------------------------------------------------
__global__ void sage_scatter(const float* __restrict__ h,
                             const int*   __restrict__ esrc,
                             const int*   __restrict__ edst,
                             const float* __restrict__ ew,
                             float*       __restrict__ neigh) {
    int gid  = blockIdx.x * blockDim.x + threadIdx.x;
    int e    = gid >> 4;                 // 16 lanes per edge
    int f0   = (gid & 15) << 2;          // feature offset: 0,4,...,60
    if (e >= N_EDGES) return;
    int   s  = esrc[e];
    int   d  = edst[e];
    float w  = ew[e];
    float4 hv = *(const float4*)(h + (size_t)s * DIM + f0);
    float* base = neigh + (size_t)d * DIM + f0;
    atomicAdd(base + 0, w * hv.x);
    atomicAdd(base + 1, w * hv.y);
    atomicAdd(base + 2, w * hv.z);
    atomicAdd(base + 3, w * hv.w);
}

// ---------------------------------------------------------------------------
// Kernel 2: out = relu(h @ W_self^T + b_self + neigh @ W_neigh^T + b_neigh)
// One wave per 16x16 output tile, V_WMMA_F32_16X16X4_F32, K=64 -> 16 steps
// per weight matrix, both GEMMs accumulated into one v8f fragment.
//
// A fragment (16x4, f32): lane<16 -> row M=lane, K={k,k+1};
//                         lane>=16 -> row M=lane-16, K={k+2,k+3}.
// B fragment (4x16): B[k][n] = W[n][k]; lane<16 -> col N=lane, K={k,k+1};
//                         lane>=16 -> col N=lane-16, K={k+2,k+3}.
// D fragment: v in 0..7: lane<16 -> (M = m0+v,  N = m0n+lane),
//                        lane>=16 -> (M = m0+8+v, N = n0+lane-16).
// ---------------------------------------------------------------------------
__global__ void sage_gemm_wmma(const float* __restrict__ h,
                               const float* __restrict__ neigh,
                               const float* __restrict__ Wself,
                               const float* __restrict__ bself,
                               const float* __restrict__ Wneigh,
                               const float* __restrict__ bneigh,
                               float*       __restrict__ out) {
    const int tilesRow = N_NODES / 16;   // 3125
    const int tilesCol = DIM / 16;       // 4

    int gtid = blockIdx.x * blockDim.x + threadIdx.x;
    int wave = gtid >> 5;
    int lane = threadIdx.x & 31;

    int rowTile = wave / tilesCol;
    int colTile = wave % tilesCol;
    if (rowTile >= tilesRow) return;     // wave-uniform: EXEC stays all-ones

    int m0 = rowTile * 16;
    int n0 = colTile * 16;

    int lane16 = lane & 15;
    int khalf  = (lane >> 4) << 1;       // 0 for lanes 0-15, 2 for lanes 16-31

    const float* hRow  = h     + (size_t)(m0 + lane16) * DIM;   // A rows
    const float* nRow  = neigh + (size_t)(m0 + lane16) * DIM;
    const float* wsRow = Wself  + (size_t)(n0 + lane16) * DIM;  // B cols (W row-major -> W^T col)
    const float* wnRow = Wneigh + (size_t)(n0 + lane16) * DIM;

    v8f acc = {};

    #pragma unroll
    for (int k = 0; k < DIM; k += 4) {
        v2f a = *(const v2f*)(hRow  + k + khalf);
        v2f b = *(const v2f*)(wsRow + k + khalf);
        acc = __builtin_amdgcn_wmma_f32_16x16x4_f32(
                  false, a, false, b, (short)0, acc, false, false);
    }
    #pragma unroll
    for (int k = 0; k < DIM; k += 4) {
        v2f a = *(const v2f*)(nRow  + k + khalf);
        v2f b = *(const v2f*)(wnRow + k + khalf);
        acc = __builtin_amdgcn_wmma_f32_16x16x4_f32(
                  false, a, false, b, (short)0, acc, false, false);
    }

    // Fused bias + ReLU + store.
    int nIdx  = n0 + lane16;
    int mBase = m0 + ((lane >> 4) << 3);  // +8 for upper half-lanes
    float bias = bself[nIdx] + bneigh[nIdx];
    #pragma unroll
    for (int v = 0; v < 8; ++v) {
        float val = acc[v] + bias;
        out[(size_t)(mBase + v) * DIM + nIdx] = val > 0.f ? val : 0.f;
    }
}

// ---------------------------------------------------------------------------
extern "C" void kernel_launch(void* const* d_in, const int* in_sizes, int n_in,
                              void* d_out, int out_size, void* d_ws, size_t ws_size,
                              hipStream_t stream) {
    const float* h      = (const float*)d_in[0];
    const int*   esrc   = (const int*)  d_in[1];
    const int*   edst   = (const int*)  d_in[2];
    const float* ew     = (const float*)d_in[3];
    const float* Wself  = (const float*)d_in[4];
    const float* bself  = (const float*)d_in[5];
    const float* Wneigh = (const float*)d_in[6];
    const float* bneigh = (const float*)d_in[7];
    float* out   = (float*)d_out;
    float* neigh = (float*)d_ws;                 // N_NODES * DIM floats = 12.8 MB

    // 0) zero accumulator
    {
        int n4 = N_NODES * DIM / 4;              // 800k float4
        int blocks = (n4 + 255) / 256;
        sage_zero_ws<<<blocks, 256, 0, stream>>>((float4*)neigh, n4);
    }
    // 1) edge scatter-add
    {
        long long threads = (long long)N_EDGES * 16;
        int blocks = (int)((threads + 255) / 256);
        sage_scatter<<<blocks, 256, 0, stream>>>(h, esrc, edst, ew, neigh);
    }
    // 2) WMMA GEMM + bias + relu
    {
        int waves = (N_NODES / 16) * (DIM / 16);  // 12500
        int blocks = (waves * 32 + 255) / 256;
        sage_gemm_wmma<<<blocks, 256, 0, stream>>>(h, neigh, Wself, bself,
                                                   Wneigh, bneigh, out);
    }
}